// FinalAttentionQKV_69106023792697
// MI455X (gfx1250) — compile-verified
//
#include <hip/hip_runtime.h>
#include <math.h>

typedef __attribute__((ext_vector_type(2))) float v2f;
typedef __attribute__((ext_vector_type(8))) float v8f;

#define NEG_INF_VAL (-1e9f)

// ---------------- K1: partial column sums over T (for mean) ----------------
__global__ void mean_partial_kernel(const float* __restrict__ x, float* __restrict__ part,
                                    int B, int T, int D, int SPLIT) {
    int b  = blockIdx.x / SPLIT;
    int sp = blockIdx.x % SPLIT;
    int d  = threadIdx.x;              // blockDim.x == D
    int tchunk = T / SPLIT;
    const float* p = x + ((size_t)b * T + (size_t)sp * tchunk) * D + d;
    float acc = 0.f;
    for (int t = 0; t < tchunk; ++t) {
        __builtin_prefetch(p + 8 * D, 0, 1);   // global_prefetch ahead of the stream
        acc += *p;
        p += D;
    }
    part[((size_t)b * SPLIT + sp) * D + d] = acc;
}

__global__ void mean_reduce_kernel(const float* __restrict__ part, float* __restrict__ mean,
                                   int T, int D, int SPLIT) {
    int b = blockIdx.x, d = threadIdx.x;
    float acc = 0.f;
    for (int sp = 0; sp < SPLIT; ++sp) acc += part[((size_t)b * SPLIT + sp) * D + d];
    mean[(size_t)b * D + d] = acc / (float)T;
}

// ---------------- WMMA f32 16x16x4 small GEMM: C = A*B(+bias[*rowscale]) ----
// One wave per 16x16 C tile. A: MxK row-major. B: KxN (or NxK if TRANSB).
// ISA layouts (CDNA5 §7.12.2):
//   A 16x4 f32: lanes 0-15 -> M=lane, VGPR0/1 = K=0/1 ; lanes 16-31 -> K=2/3
//   B 4x16 f32: mirrored (N across lanes, K across VGPRs/lane-halves)
//   C 16x16 f32: VGPR r -> M = r + 8*(lane>=16), N = lane&15
template <bool TRANSB, bool ADD_BIAS, bool ROWSCALE>
__global__ void __launch_bounds__(32) wmma_gemm16(const float* __restrict__ A, int lda,
                                                  const float* __restrict__ Bm, int ldb,
                                                  const float* __restrict__ bias,
                                                  const float* __restrict__ rowscale,
                                                  float* __restrict__ C, int ldc,
                                                  int N, int K) {
    int ntiles = N >> 4;
    int tm = blockIdx.x / ntiles;
    int tn = blockIdx.x % ntiles;
    int lane = threadIdx.x & 31;
    int mr   = lane & 15;
    int koff = (lane >> 4) << 1;                    // 0 or 2
    const float* Arow = A + (size_t)(tm * 16 + mr) * lda;
    int gn = tn * 16 + mr;                          // N index for this lane
    v8f c = {0.f, 0.f, 0.f, 0.f, 0.f, 0.f, 0.f, 0.f};
    for (int k0 = 0; k0 < K; k0 += 4) {
        v2f a, b;
        a.x = Arow[k0 + koff];
        a.y = Arow[k0 + koff + 1];
        if (TRANSB) {
            b.x = Bm[(size_t)gn * ldb + k0 + koff];
            b.y = Bm[(size_t)gn * ldb + k0 + koff + 1];
        } else {
            b.x = Bm[(size_t)(k0 + koff) * ldb + gn];
            b.y = Bm[(size_t)(k0 + koff + 1) * ldb + gn];
        }
        c = __builtin_amdgcn_wmma_f32_16x16x4_f32(false, a, false, b, (short)0, c,
                                                  false, false);
    }
    int mbase = tm * 16 + ((lane >> 4) << 3);
#pragma unroll
    for (int r = 0; r < 8; ++r) {
        int gm = mbase + r;
        float v = c[r];
        if (ADD_BIAS) {
            float bb = bias[gn];
            if (ROWSCALE) bb *= rowscale[gm];
            v += bb;
        }
        C[(size_t)gm * ldc + gn] = v;
    }
}

// ---------------- cb[b] = dot(bk, q[b,:]) ----------------
__global__ void rowdot_kernel(const float* __restrict__ q, const float* __restrict__ bk,
                              float* __restrict__ cb, int H) {
    __shared__ float red[256];
    int b = blockIdx.x, h = threadIdx.x;
    red[h] = (h < H) ? q[(size_t)b * H + h] * bk[h] : 0.f;
    __syncthreads();
    for (int s = 128; s > 0; s >>= 1) {
        if (h < s) red[h] += red[h + s];
        __syncthreads();
    }
    if (h == 0) cb[b] = red[0];
}

// ---------------- K3: scores e[b,t] = x[b,t,:]·wkq[b,:] + cb, then gating ---
__global__ void scores_kernel(const float* __restrict__ x, const float* __restrict__ wkq,
                              const float* __restrict__ cb, const int* __restrict__ mask,
                              const float* __restrict__ rate, float* __restrict__ e,
                              int T, int D, int ROWS) {
    int chunks = T / ROWS;
    int b  = blockIdx.x / chunks;
    int t0 = (blockIdx.x % chunks) * ROWS;
    int lane = threadIdx.x & 31;
    int wid  = threadIdx.x >> 5;
    int nw   = blockDim.x >> 5;
    // D == 256: each lane caches 8 wkq values
    const float4* wp = (const float4*)(wkq + (size_t)b * D + lane * 8);
    float4 w0 = wp[0], w1 = wp[1];
    float cbb = cb[b];
    for (int r = wid; r < ROWS; r += nw) {
        int t = t0 + r;
        const float4* xp = (const float4*)(x + ((size_t)b * T + t) * D + lane * 8);
        __builtin_prefetch(xp + 2 * nw * (D / 4) / 2, 0, 1);
        float4 x0 = xp[0], x1 = xp[1];
        float dot = x0.x * w0.x + x0.y * w0.y + x0.z * w0.z + x0.w * w0.w
                  + x1.x * w1.x + x1.y * w1.y + x1.z * w1.z + x1.w * w1.w;
        for (int off = 16; off > 0; off >>= 1) dot += __shfl_xor(dot, off, 32);
        if (lane == 0) {
            float ev = dot + cbb;
            int   mk = mask[(size_t)b * T + t];
            float sig = 1.f / (1.f + expf(-ev));
            float tmiss = log1pf((1.f - sig) * (float)mk);
            ev = ev - rate[t] * tmiss;
            if (mk == 0) ev = NEG_INF_VAL;
            e[(size_t)b * T + t] = ev;
        }
    }
}

// ---------------- K4: sparsemax per row (T=4096), bitonic sort in LDS -------
__global__ void __launch_bounds__(256) sparsemax_kernel(const float* __restrict__ e,
                                                        float* __restrict__ a,
                                                        float* __restrict__ sumA, int T) {
    const int N = 4096, NT = 256, PER = N / NT; // 16 elems/thread
    __shared__ float z[N];
    __shared__ float zs[N];
    __shared__ float cs[N];
    __shared__ float red[NT];
    int b = blockIdx.x, tid = threadIdx.x;

    // load + global max
    float mx = -3.4e38f;
    for (int i = tid; i < N; i += NT) {
        float v = e[(size_t)b * T + i];
        z[i] = v;
        mx = fmaxf(mx, v);
    }
    red[tid] = mx; __syncthreads();
    for (int s = 128; s > 0; s >>= 1) { if (tid < s) red[tid] = fmaxf(red[tid], red[tid + s]); __syncthreads(); }
    mx = red[0]; __syncthreads();
    for (int i = tid; i < N; i += NT) { float v = z[i] - mx; z[i] = v; zs[i] = v; }
    __syncthreads();

    // bitonic sort descending
    for (int ksz = 2; ksz <= N; ksz <<= 1) {
        for (int jsz = ksz >> 1; jsz > 0; jsz >>= 1) {
            for (int i = tid; i < N; i += NT) {
                int ixj = i ^ jsz;
                if (ixj > i) {
                    float va = zs[i], vb = zs[ixj];
                    bool desc = ((i & ksz) == 0);
                    if (desc ? (va < vb) : (va > vb)) { zs[i] = vb; zs[ixj] = va; }
                }
            }
            __syncthreads();
        }
    }

    // inclusive cumsum: local scan + scan of block sums
    {
        int base = tid * PER;
        float run = 0.f;
        for (int j = 0; j < PER; ++j) { run += zs[base + j]; cs[base + j] = run; }
        red[tid] = run; __syncthreads();
        for (int off = 1; off < NT; off <<= 1) {
            float v = (tid >= off) ? red[tid - off] : 0.f;
            __syncthreads();
            red[tid] += v;
            __syncthreads();
        }
        float pre = (tid > 0) ? red[tid - 1] : 0.f;
        for (int j = 0; j < PER; ++j) cs[base + j] += pre;
    }
    __syncthreads();

    // k = max(is_gt * r), ssel = sum(is_gt * zs)
    float kloc = 0.f, sloc = 0.f;
    {
        int base = tid * PER;
        for (int j = 0; j < PER; ++j) {
            int idx = base + j;
            float r  = (float)(idx + 1);
            float zz = zs[idx];
            if (1.0f + r * zz > cs[idx]) { kloc = fmaxf(kloc, r); sloc += zz; }
        }
    }
    red[tid] = kloc; __syncthreads();
    for (int s = 128; s > 0; s >>= 1) { if (tid < s) red[tid] = fmaxf(red[tid], red[tid + s]); __syncthreads(); }
    float kk = red[0]; __syncthreads();
    red[tid] = sloc; __syncthreads();
    for (int s = 128; s > 0; s >>= 1) { if (tid < s) red[tid] += red[tid + s]; __syncthreads(); }
    float ssum = red[0]; __syncthreads();
    float tau = (ssum - 1.0f) / kk;

    // a = max(0, z - tau); also sum(a) for the bias term of out
    float saloc = 0.f;
    for (int i = tid; i < N; i += NT) {
        float av = fmaxf(0.f, z[i] - tau);
        a[(size_t)b * T + i] = av;
        saloc += av;
    }
    red[tid] = saloc; __syncthreads();
    for (int s = 128; s > 0; s >>= 1) { if (tid < s) red[tid] += red[tid + s]; __syncthreads(); }
    if (tid == 0) sumA[b] = red[0];
}

// ---------------- K5: s[b,d] = sum_t a[b,t]*x[b,t,d] (split-T partials) -----
__global__ void wsum_partial_kernel(const float* __restrict__ x, const float* __restrict__ a,
                                    float* __restrict__ part, int T, int D, int SPLIT) {
    int b  = blockIdx.x / SPLIT;
    int sp = blockIdx.x % SPLIT;
    int d  = threadIdx.x;
    int tchunk = T / SPLIT;
    const float* p  = x + ((size_t)b * T + (size_t)sp * tchunk) * D + d;
    const float* ap = a + (size_t)b * T + (size_t)sp * tchunk;
    float acc = 0.f;
    for (int t = 0; t < tchunk; ++t) {
        __builtin_prefetch(p + 8 * D, 0, 1);
        acc += ap[t] * p[0];
        p += D;
    }
    part[((size_t)b * SPLIT + sp) * D + d] = acc;
}

__global__ void wsum_reduce_kernel(const float* __restrict__ part, float* __restrict__ s,
                                   int D, int SPLIT) {
    int b = blockIdx.x, d = threadIdx.x;
    float acc = 0.f;
    for (int sp = 0; sp < SPLIT; ++sp) acc += part[((size_t)b * SPLIT + sp) * D + d];
    s[(size_t)b * D + d] = acc;
}

// ---------------- launcher ----------------
extern "C" void kernel_launch(void* const* d_in, const int* in_sizes, int n_in,
                              void* d_out, int out_size, void* d_ws, size_t ws_size,
                              hipStream_t stream) {
    const float* input = (const float*)d_in[0];
    const int*   mask  = (const int*)d_in[1];
    const float* Wq    = (const float*)d_in[2];
    const float* bq    = (const float*)d_in[3];
    const float* Wk    = (const float*)d_in[4];
    const float* bk    = (const float*)d_in[5];
    const float* Wv    = (const float*)d_in[6];
    const float* bv    = (const float*)d_in[7];
    const float* rate  = (const float*)d_in[8];

    const int T = in_sizes[8];                 // 4096
    const int B = in_sizes[1] / T;             // 64
    const int D = in_sizes[0] / (B * T);       // 256
    const int H = in_sizes[2] / D;             // 256
    const int SPLIT = 16;
    const int ROWS  = 64;

    float* ws = (float*)d_ws;
    float* mean_part = ws;                 ws += (size_t)B * SPLIT * D;
    float* mean      = ws;                 ws += (size_t)B * D;
    float* q         = ws;                 ws += (size_t)B * H;
    float* wkq       = ws;                 ws += (size_t)B * D;
    float* cb        = ws;                 ws += (size_t)B;
    float* e         = ws;                 ws += (size_t)B * T;
    float* sumA      = ws;                 ws += (size_t)B;
    float* s_part    = ws;                 ws += (size_t)B * SPLIT * D;
    float* s         = ws;                 ws += (size_t)B * D;

    float* out_ptr = (float*)d_out;            // (B,H)
    float* a_ptr   = out_ptr + (size_t)B * H;  // (B,T)

    // 1) mean over T
    mean_partial_kernel<<<B * SPLIT, D, 0, stream>>>(input, mean_part, B, T, D, SPLIT);
    mean_reduce_kernel<<<B, D, 0, stream>>>(mean_part, mean, T, D, SPLIT);

    // 2) q = mean @ Wq + bq           (WMMA f32)
    wmma_gemm16<false, true, false><<<(B / 16) * (H / 16), 32, 0, stream>>>(
        mean, D, Wq, H, bq, nullptr, q, H, H, D);

    // 3) wkq = q @ Wk^T               (WMMA f32) ; cb = bk . q
    wmma_gemm16<true, false, false><<<(B / 16) * (D / 16), 32, 0, stream>>>(
        q, H, Wk, H, nullptr, nullptr, wkq, D, D, H);
    rowdot_kernel<<<B, 256, 0, stream>>>(q, bk, cb, H);

    // 4) e[b,t] = x . wkq + cb, gated by sigmoid/log1p/mask/rate
    scores_kernel<<<B * (T / ROWS), 256, 0, stream>>>(input, wkq, cb, mask, rate, e,
                                                      T, D, ROWS);

    // 5) sparsemax per row -> a (second output) and sumA
    sparsemax_kernel<<<B, 256, 0, stream>>>(e, a_ptr, sumA, T);

    // 6) s[b,:] = sum_t a * x
    wsum_partial_kernel<<<B * SPLIT, D, 0, stream>>>(input, a_ptr, s_part, T, D, SPLIT);
    wsum_reduce_kernel<<<B, D, 0, stream>>>(s_part, s, D, SPLIT);

    // 7) out = s @ Wv + sumA * bv     (WMMA f32)
    wmma_gemm16<false, true, true><<<(B / 16) * (H / 16), 32, 0, stream>>>(
        s, D, Wv, H, bv, sumA, out_ptr, H, H, D);
}